// MyGCN_79946521247768
// MI455X (gfx1250) — compile-verified
//
#include <hip/hip_runtime.h>
#include <math.h>

#define NN 51200
#define BB 256
#define EE 819200
#define NROI 200

typedef float v2f __attribute__((ext_vector_type(2)));
typedef float v8f __attribute__((ext_vector_type(8)));

// ---------------------------------------------------------------------------
// WMMA fp32 GEMM:  C[:, coloff:coloff+Nc] = act(A @ W + bias)
// A: [M,*] row-major with leading dim lda, column offset aoff, logical K cols
// W: [K,Nc] row-major compact.  M must be a multiple of 128.
// Block: 256 threads = 8 waves; tile 128 (M) x 16 (N); K chunked by 64 in LDS.
// ---------------------------------------------------------------------------
#define TM 128
#define TK 64

__global__ __launch_bounds__(256) void gemm_wmma(
    const float* __restrict__ A, int lda, int aoff,
    const float* __restrict__ W, const float* __restrict__ bias,
    float* __restrict__ C, int M, int K, int Nc, int ldc, int coloff, int relu)
{
    __shared__ float sA[TM * TK];   // 32 KB
    __shared__ float sW[TK * 16];   //  4 KB
    const int tid  = threadIdx.x;
    const int wave = tid >> 5;
    const int lane = tid & 31;
    const int m0 = blockIdx.x * TM;
    const int n0 = blockIdx.y * 16;

    v8f acc = {};
    const int nk = (K + TK - 1) / TK;
    const int colL  = lane & 15;
    const int kpair = (lane >> 4) << 1;
    const int arow  = (wave << 4) + colL;   // row within 128-row tile for A operand

    for (int c = 0; c < nk; ++c) {
        const int k0 = c * TK;
        // stage A tile (coalesced: consecutive tid -> consecutive k)
        for (int i = tid; i < TM * TK; i += 256) {
            int r  = i >> 6;
            int kk = i & (TK - 1);
            int gk = k0 + kk;
            sA[i] = (gk < K) ? A[(size_t)(m0 + r) * lda + aoff + gk] : 0.0f;
        }
        // stage W strip
        for (int i = tid; i < TK * 16; i += 256) {
            int kr = i >> 4;
            int cc = i & 15;
            int gk = k0 + kr;
            int gc = n0 + cc;
            sW[i] = (gk < K && gc < Nc) ? W[(size_t)gk * Nc + gc] : 0.0f;
        }
        __syncthreads();
        // 16 WMMA steps of K=4 each
        for (int kk = 0; kk < TK; kk += 4) {
            v2f a, b;
            a.x = sA[arow * TK + kk + kpair];
            a.y = sA[arow * TK + kk + kpair + 1];
            b.x = sW[(kk + kpair) * 16 + colL];
            b.y = sW[(kk + kpair + 1) * 16 + colL];
            acc = __builtin_amdgcn_wmma_f32_16x16x4_f32(
                false, a, false, b, (short)0, acc, false, false);
        }
        __syncthreads();
    }

    const int gc = n0 + colL;
    if (gc < Nc) {
        const float bval = bias ? bias[gc] : 0.0f;
        const int rbase = m0 + (wave << 4) + ((lane >> 4) << 3);
        for (int r = 0; r < 8; ++r) {
            float v = acc[r] + bval;
            if (relu) v = fmaxf(v, 0.0f);
            C[(size_t)(rbase + r) * ldc + coloff + gc] = v;
        }
    }
}

// ---------------------------------------------------------------------------
// elementwise / graph kernels
// ---------------------------------------------------------------------------
__global__ void fill_ker(float* p, float v, int n) {
    int i = blockIdx.x * blockDim.x + threadIdx.x;
    if (i < n) p[i] = v;
}

__global__ void deg_edges_ker(const int* __restrict__ ei, const float* __restrict__ ew,
                              float* __restrict__ deg) {
    int e = blockIdx.x * blockDim.x + threadIdx.x;
    if (e < EE) atomicAdd(&deg[ei[EE + e]], ew[e]);
}

__global__ void rsqrt_ker(float* d) {
    int i = blockIdx.x * blockDim.x + threadIdx.x;
    if (i < NN) { float v = d[i]; d[i] = (v > 0.0f) ? rsqrtf(v) : 0.0f; }
}

// out[n, off+4g..] = h[n,4g..] * dinv[n]^2 + bias[4g..]   (self-loop + bias), float4
__global__ void gcn_init_ker(const float* __restrict__ h, const float* __restrict__ dinv,
                             const float* __restrict__ bias, float* __restrict__ out,
                             int Fq, int logFq, int ld, int off) {
    int idx = blockIdx.x * blockDim.x + threadIdx.x;
    if (idx >= NN * Fq) return;
    int g = idx & (Fq - 1);
    int n = idx >> logFq;
    float di = dinv[n];
    float c  = di * di;
    float4 hv = ((const float4*)h)[(size_t)n * Fq + g];
    float4 bv = ((const float4*)bias)[g];
    float4 r;
    r.x = hv.x * c + bv.x;
    r.y = hv.y * c + bv.y;
    r.z = hv.z * c + bv.z;
    r.w = hv.w * c + bv.w;
    *(float4*)(out + (size_t)n * ld + off + 4 * g) = r;
}

// out[dst, off+4g..] += h[src,4g..] * dinv[src]*w*dinv[dst]   (float4 gather, 4 atomics)
__global__ void gcn_edges_ker(const int* __restrict__ ei, const float* __restrict__ ew,
                              const float* __restrict__ dinv, const float* __restrict__ h,
                              float* __restrict__ out, int Fq, int logFq, int ld, int off) {
    int idx = blockIdx.x * blockDim.x + threadIdx.x;
    if (idx >= EE * Fq) return;
    int g = idx & (Fq - 1);
    int e = idx >> logFq;
    int s = ei[e];
    int d = ei[EE + e];
    float coef = dinv[s] * ew[e] * dinv[d];
    float4 hv = ((const float4*)h)[(size_t)s * Fq + g];
    float* po = out + (size_t)d * ld + off + 4 * g;
    atomicAdd(po + 0, hv.x * coef);
    atomicAdd(po + 1, hv.y * coef);
    atomicAdd(po + 2, hv.z * coef);
    atomicAdd(po + 3, hv.w * coef);
}

__global__ void relu_ker(float* __restrict__ out, int Fq, int logFq, int ld, int off) {
    int idx = blockIdx.x * blockDim.x + threadIdx.x;
    if (idx >= NN * Fq) return;
    int g = idx & (Fq - 1);
    int n = idx >> logFq;
    float4* p = (float4*)(out + (size_t)n * ld + off + 4 * g);
    float4 v = *p;
    v.x = fmaxf(v.x, 0.0f); v.y = fmaxf(v.y, 0.0f);
    v.z = fmaxf(v.z, 0.0f); v.w = fmaxf(v.w, 0.0f);
    *p = v;
}

// cat([segment_max, segment_mean]) over contiguous 200-row segments
__global__ void pool_ker(const float* __restrict__ x, int F,
                         float* __restrict__ out, int ldo, int off) {
    int b = blockIdx.x;
    for (int f = threadIdx.x; f < F; f += blockDim.x) {
        float mx = -3.402823e38f, sm = 0.0f;
        const float* base = x + (size_t)b * NROI * F + f;
        for (int i = 0; i < NROI; ++i) {
            float v = base[(size_t)i * F];
            mx = fmaxf(mx, v);
            sm += v;
        }
        out[(size_t)b * ldo + off + f]     = mx;
        out[(size_t)b * ldo + off + F + f] = sm * (1.0f / (float)NROI);
    }
}

// batchnorm over B=256 rows of column d (grid = D columns, block = 256)
__global__ __launch_bounds__(256) void bn_ker(
    const float* __restrict__ x, int ldx, int offx,
    const float* __restrict__ g, const float* __restrict__ bb,
    float* __restrict__ out, int ldo, int offo) {
    __shared__ float s1[256];
    __shared__ float s2[256];
    int d = blockIdx.x;
    int t = threadIdx.x;
    float v = x[(size_t)t * ldx + offx + d];
    s1[t] = v; s2[t] = v * v;
    __syncthreads();
    for (int st = 128; st > 0; st >>= 1) {
        if (t < st) { s1[t] += s1[t + st]; s2[t] += s2[t + st]; }
        __syncthreads();
    }
    float m   = s1[0] * (1.0f / 256.0f);
    float var = s2[0] * (1.0f / 256.0f) - m * m;
    float sc  = g[d] * rsqrtf(var + 1e-5f);
    out[(size_t)t * ldo + offo + d] = (v - m) * sc + bb[d];
}

// dst[n, offd+f] = (n%2==p && f%2==p) ? src[n, offs+f] : 0  (float4: col parity = lane in vec)
__global__ void split_ker(const float* __restrict__ src, int lds_, int offs,
                          float* __restrict__ dst, int ldd, int offd,
                          int Fq, int logFq, int parity) {
    int idx = blockIdx.x * blockDim.x + threadIdx.x;
    if (idx >= NN * Fq) return;
    int g = idx & (Fq - 1);
    int n = idx >> logFq;
    float4 v = ((const float4*)(src + (size_t)n * lds_ + offs))[g];
    bool rowsel = ((n & 1) == parity);
    float4 o;
    if (parity == 0) {   // keep even rows, even cols (vec lanes 0,2)
        o.x = rowsel ? v.x : 0.0f; o.y = 0.0f;
        o.z = rowsel ? v.z : 0.0f; o.w = 0.0f;
    } else {             // keep odd rows, odd cols (vec lanes 1,3)
        o.x = 0.0f; o.y = rowsel ? v.y : 0.0f;
        o.z = 0.0f; o.w = rowsel ? v.w : 0.0f;
    }
    *(float4*)(dst + (size_t)n * ldd + offd + 4 * g) = o;
}

// acc += sum_n sqrt(sum_f (a-b+1e-6)^2)  -- one wave per row, coalesced
__global__ void pdist_ker(const float* __restrict__ a, int lda, int aoff,
                          const float* __restrict__ b, int ldb, int boff,
                          int F, int rows, float* __restrict__ acc) {
    int w    = (blockIdx.x * blockDim.x + threadIdx.x) >> 5;
    int lane = threadIdx.x & 31;
    if (w >= rows) return;
    float s = 0.0f;
    for (int f = lane; f < F; f += 32) {
        float d = a[(size_t)w * lda + aoff + f] - b[(size_t)w * ldb + boff + f] + 1e-6f;
        s += d * d;
    }
    for (int o = 16; o > 0; o >>= 1) s += __shfl_xor(s, o, 32);
    if (lane == 0) atomicAdd(acc, sqrtf(s));
}

__global__ void losses_ker(const float* __restrict__ acc, float* __restrict__ out) {
    if (blockIdx.x == 0 && threadIdx.x == 0) {
        float d1 = acc[0] * (1.0f / (float)BB);
        float d2 = acc[1] * (1.0f / (float)BB);
        out[512] = 0.5f * (d1 + d2);
        out[513] = (acc[2] + acc[3] + acc[4] + acc[5]) * (1.0f / (4.0f * (float)NN));
    }
}

__global__ void logsoftmax2_ker(const float* __restrict__ logits, float* __restrict__ out) {
    int i = blockIdx.x * blockDim.x + threadIdx.x;
    if (i >= BB) return;
    float a = logits[2 * i], b = logits[2 * i + 1];
    float m = fmaxf(a, b);
    float lse = m + logf(expf(a - m) + expf(b - m));
    out[2 * i]     = a - lse;
    out[2 * i + 1] = b - lse;
}

// ---------------------------------------------------------------------------
// host-side helpers
// ---------------------------------------------------------------------------
static inline void gemm(hipStream_t s, const float* A, int lda, int aoff,
                        const float* W, const float* bias, float* C,
                        int M, int K, int Nc, int ldc, int coloff, int relu) {
    dim3 g(M / TM, (Nc + 15) / 16);
    gemm_wmma<<<g, 256, 0, s>>>(A, lda, aoff, W, bias, C, M, K, Nc, ldc, coloff, relu);
}

static inline void make_dinv(hipStream_t s, const int* ei, const float* ew, float* dinv) {
    fill_ker<<<(NN + 255) / 256, 256, 0, s>>>(dinv, 1.0f, NN);          // self-loop weight
    deg_edges_ker<<<(EE + 255) / 256, 256, 0, s>>>(ei, ew, dinv);
    rsqrt_ker<<<(NN + 255) / 256, 256, 0, s>>>(dinv);
}

static inline void run_gcn(hipStream_t s, const float* X, int lda, int aoff, int inF,
                           const float* Wt, const float* bias, int outF, int logF,
                           const int* ei, const float* ew, const float* dinv,
                           float* htmp, float* out, int ld, int off, int relu) {
    gemm(s, X, lda, aoff, Wt, nullptr, htmp, NN, inF, outF, outF, 0, 0);
    const int Fq = outF >> 2, logFq = logF - 2;
    int tot = NN * Fq;
    gcn_init_ker<<<(tot + 255) / 256, 256, 0, s>>>(htmp, dinv, bias, out, Fq, logFq, ld, off);
    int tote = EE * Fq;
    gcn_edges_ker<<<(tote + 255) / 256, 256, 0, s>>>(ei, ew, dinv, htmp, out, Fq, logFq, ld, off);
    if (relu) relu_ker<<<(tot + 255) / 256, 256, 0, s>>>(out, Fq, logFq, ld, off);
}

extern "C" void kernel_launch(void* const* d_in, const int* in_sizes, int n_in,
                              void* d_out, int out_size, void* d_ws, size_t ws_size,
                              hipStream_t stream) {
    // ---- inputs (setup_inputs dict order; params dict flattened in insertion order)
    const float* x      = (const float*)d_in[0];
    const int*   ei_m   = (const int*)  d_in[1];
    const float* ew_m   = (const float*)d_in[3];
    const float* upper  = (const float*)d_in[4];
    const float* lx_in  = (const float*)d_in[5];
    const int*   ei_l   = (const int*)  d_in[6];
    const float* ew_l   = (const float*)d_in[7];
    const float* rx_in  = (const float*)d_in[8];
    const int*   ei_r   = (const int*)  d_in[9];
    const float* ew_r   = (const float*)d_in[10];
    const float* bx_in  = (const float*)d_in[11];
    const int*   ei_b   = (const int*)  d_in[12];
    const float* ew_b   = (const float*)d_in[13];
    int p = 14;
    const float* conv1_W = (const float*)d_in[p++]; const float* conv1_b = (const float*)d_in[p++];
    const float* conv2_W = (const float*)d_in[p++]; const float* conv2_b = (const float*)d_in[p++];
    const float* u_fc1_W = (const float*)d_in[p++]; const float* u_fc1_b = (const float*)d_in[p++];
    const float* u_fc2_W = (const float*)d_in[p++]; const float* u_fc2_b = (const float*)d_in[p++];
    p += 2; // u_fc3_W, u_fc3_b unused by forward
    const float* u_bn1_g = (const float*)d_in[p++]; const float* u_bn1_b = (const float*)d_in[p++];
    const float* u_bn2_g = (const float*)d_in[p++]; const float* u_bn2_b = (const float*)d_in[p++];
    const float* m_fc1_W = (const float*)d_in[p++]; const float* m_fc1_b = (const float*)d_in[p++];
    const float* m_fc2_W = (const float*)d_in[p++]; const float* m_fc2_b = (const float*)d_in[p++];
    const float* m_bn0_g = (const float*)d_in[p++]; const float* m_bn0_b = (const float*)d_in[p++];
    const float* m_bn1_g = (const float*)d_in[p++]; const float* m_bn1_b = (const float*)d_in[p++];
    const float* cls_W   = (const float*)d_in[p++]; const float* cls_b   = (const float*)d_in[p++];
    const float* h_l1_W  = (const float*)d_in[p++]; const float* h_l1_b  = (const float*)d_in[p++];
    const float* h_l2_W  = (const float*)d_in[p++]; const float* h_l2_b  = (const float*)d_in[p++];
    const float* h_l3_W  = (const float*)d_in[p++]; const float* h_l3_b  = (const float*)d_in[p++];
    const float* h_r1_W  = (const float*)d_in[p++]; const float* h_r1_b  = (const float*)d_in[p++];
    const float* h_r2_W  = (const float*)d_in[p++]; const float* h_r2_b  = (const float*)d_in[p++];
    const float* h_r3_W  = (const float*)d_in[p++]; const float* h_r3_b  = (const float*)d_in[p++];
    const float* h_b1_W  = (const float*)d_in[p++]; const float* h_b1_b  = (const float*)d_in[p++];
    const float* h_b2_W  = (const float*)d_in[p++]; const float* h_b2_b  = (const float*)d_in[p++];
    const float* h_fc_W  = (const float*)d_in[p++]; const float* h_fc_b  = (const float*)d_in[p++];
    const float* h_bn_g  = (const float*)d_in[p++]; const float* h_bn_b  = (const float*)d_in[p++];

    float* out = (float*)d_out;   // [cls 512 | closs 1 | hloss 1 | m 8192 | logits 512]
    float* m_out      = out + 514;
    float* logits_out = out + 8706;

    // ---- workspace layout (floats; all offsets 16B-aligned)
    float* ws = (float*)d_ws;
    size_t o = 0;
    float* dinv_m = ws + o; o += NN;
    float* dinv_l = ws + o; o += NN;
    float* dinv_r = ws + o; o += NN;
    float* dinv_b = ws + o; o += NN;
    float* htmp   = ws + o; o += (size_t)NN * 64;
    float* X1     = ws + o; o += (size_t)NN * 64;
    float* X2     = ws + o; o += (size_t)NN * 16;
    float* BX1    = ws + o; o += (size_t)NN * 64;
    float* LXCAT  = ws + o; o += (size_t)NN * 128;   // [lx1 | lxb]
    float* RXCAT  = ws + o; o += (size_t)NN * 128;   // [rx1 | rxb]
    float* BXCAT  = ws + o; o += (size_t)NN * 64;    // [lx2 | rx2]
    float* BX2    = ws + o; o += (size_t)NN * 16;
    float* LX3    = ws + o; o += (size_t)NN * 16;
    float* RX3    = ws + o; o += (size_t)NN * 16;
    float* LXB2   = ws + o; o += (size_t)NN * 16;
    float* RXB2   = ws + o; o += (size_t)NN * 16;
    float* XFEAT  = ws + o; o += (size_t)BB * 352;   // [xr1 128 | xr2 32 | upper 128 | harf 64]
    float* FE96   = ws + o; o += (size_t)BB * 96;    // [ro2 | lro | rro]
    float* RO1    = ws + o; o += (size_t)BB * 128;
    float* U1     = ws + o; o += (size_t)BB * 512;
    float* U1B    = ws + o; o += (size_t)BB * 512;
    float* U2     = ws + o; o += (size_t)BB * 128;
    float* MB0    = ws + o; o += (size_t)BB * 352;
    float* M1     = ws + o; o += (size_t)BB * 256;
    float* MB1    = ws + o; o += (size_t)BB * 256;
    float* HFP    = ws + o; o += (size_t)BB * 64;
    float* ACC    = ws + o; o += 8;
    (void)ws_size; (void)in_sizes; (void)n_in; (void)out_size;

    hipStream_t s = stream;
    fill_ker<<<1, 32, 0, s>>>(ACC, 0.0f, 8);

    // degree normalizations
    make_dinv(s, ei_m, ew_m, dinv_m);
    make_dinv(s, ei_l, ew_l, dinv_l);
    make_dinv(s, ei_r, ew_r, dinv_r);
    make_dinv(s, ei_b, ew_b, dinv_b);

    const int TPB = 256;
    #define GRID(n) (((n) + TPB - 1) / TPB)

    // ---- Harf bipartite network, layer 1
    run_gcn(s, lx_in, 256, 0, 256, h_l1_W, h_l1_b, 64, 6, ei_l, ew_l, dinv_l, htmp, LXCAT, 128, 0, 1);
    run_gcn(s, rx_in, 256, 0, 256, h_r1_W, h_r1_b, 64, 6, ei_r, ew_r, dinv_r, htmp, RXCAT, 128, 0, 1);
    run_gcn(s, bx_in, 256, 0, 256, h_b1_W, h_b1_b, 64, 6, ei_b, ew_b, dinv_b, htmp, BX1, 64, 0, 1);
    pool_ker<<<BB, 64, 0, s>>>(BX1, 64, RO1, 128, 0);                                 // ro1
    split_ker<<<GRID(NN * 16), TPB, 0, s>>>(BX1, 64, 0, LXCAT, 128, 64, 16, 4, 0);    // lxb
    split_ker<<<GRID(NN * 16), TPB, 0, s>>>(BX1, 64, 0, RXCAT, 128, 64, 16, 4, 1);    // rxb
    pdist_ker<<<GRID(NN * 32), TPB, 0, s>>>(LXCAT, 128, 64, LXCAT, 128, 0, 64, NN, ACC + 2); // lc1
    pdist_ker<<<GRID(NN * 32), TPB, 0, s>>>(RXCAT, 128, 64, RXCAT, 128, 0, 64, NN, ACC + 3); // rc1

    // layer 2
    run_gcn(s, LXCAT, 128, 0, 128, h_l2_W, h_l2_b, 32, 5, ei_l, ew_l, dinv_l, htmp, BXCAT, 64, 0, 1);
    run_gcn(s, RXCAT, 128, 0, 128, h_r2_W, h_r2_b, 32, 5, ei_r, ew_r, dinv_r, htmp, BXCAT, 64, 32, 1);
    run_gcn(s, BXCAT, 64, 0, 64, h_b2_W, h_b2_b, 16, 4, ei_b, ew_b, dinv_b, htmp, BX2, 16, 0, 0);
    pool_ker<<<BB, 64, 0, s>>>(BX2, 16, FE96, 96, 0);                                 // ro2
    split_ker<<<GRID(NN * 4), TPB, 0, s>>>(BX2, 16, 0, LXB2, 16, 0, 4, 2, 0);
    split_ker<<<GRID(NN * 4), TPB, 0, s>>>(BX2, 16, 0, RXB2, 16, 0, 4, 2, 1);

    // layer 3
    run_gcn(s, BXCAT, 64, 0, 32, h_l3_W, h_l3_b, 16, 4, ei_l, ew_l, dinv_l, htmp, LX3, 16, 0, 0);
    pool_ker<<<BB, 64, 0, s>>>(LX3, 16, FE96, 96, 32);                                // lro
    run_gcn(s, BXCAT, 64, 32, 32, h_r3_W, h_r3_b, 16, 4, ei_r, ew_r, dinv_r, htmp, RX3, 16, 0, 0);
    pool_ker<<<BB, 64, 0, s>>>(RX3, 16, FE96, 96, 64);                                // rro
    pdist_ker<<<GRID(NN * 32), TPB, 0, s>>>(LXB2, 16, 0, LX3, 16, 0, 16, NN, ACC + 4); // lc2
    pdist_ker<<<GRID(NN * 32), TPB, 0, s>>>(RXB2, 16, 0, RX3, 16, 0, 16, NN, ACC + 5); // rc2

    // harf_feats = bn(relu(feats @ h_fc_W + b)) -> x_feat[:, 288:352]
    gemm(s, FE96, 96, 0, h_fc_W, h_fc_b, HFP, BB, 96, 64, 64, 0, 1);
    bn_ker<<<64, 256, 0, s>>>(HFP, 64, 0, h_bn_g, h_bn_b, XFEAT, 352, 288);

    // ---- upper MLP -> x_feat[:, 160:288]
    gemm(s, upper, 1000, 0, u_fc1_W, u_fc1_b, U1, BB, 1000, 512, 512, 0, 1);
    bn_ker<<<512, 256, 0, s>>>(U1, 512, 0, u_bn1_g, u_bn1_b, U1B, 512, 0);
    gemm(s, U1B, 512, 0, u_fc2_W, u_fc2_b, U2, BB, 512, 128, 128, 0, 1);
    bn_ker<<<128, 256, 0, s>>>(U2, 128, 0, u_bn2_g, u_bn2_b, XFEAT, 352, 160);

    // ---- main GCN path -> x_feat[:, 0:160]
    run_gcn(s, x, 256, 0, 256, conv1_W, conv1_b, 64, 6, ei_m, ew_m, dinv_m, htmp, X1, 64, 0, 1);
    pool_ker<<<BB, 64, 0, s>>>(X1, 64, XFEAT, 352, 0);                                // xr1
    run_gcn(s, X1, 64, 0, 64, conv2_W, conv2_b, 16, 4, ei_m, ew_m, dinv_m, htmp, X2, 16, 0, 0);
    pool_ker<<<BB, 64, 0, s>>>(X2, 16, XFEAT, 352, 128);                              // xr2

    // consistency distances
    pdist_ker<<<GRID(BB * 32), TPB, 0, s>>>(RO1, 128, 0, XFEAT, 352, 0, 128, BB, ACC + 0); // d1
    pdist_ker<<<GRID(BB * 32), TPB, 0, s>>>(FE96, 96, 0, FE96, 96, 0, 32, BB, ACC + 1);    // d2

    // ---- classifier MLP
    bn_ker<<<352, 256, 0, s>>>(XFEAT, 352, 0, m_bn0_g, m_bn0_b, MB0, 352, 0);
    gemm(s, MB0, 352, 0, m_fc1_W, m_fc1_b, M1, BB, 352, 256, 256, 0, 1);
    bn_ker<<<256, 256, 0, s>>>(M1, 256, 0, m_bn1_g, m_bn1_b, MB1, 256, 0);
    gemm(s, MB1, 256, 0, m_fc2_W, m_fc2_b, m_out, BB, 256, 32, 32, 0, 1);   // m -> d_out
    gemm(s, m_out, 32, 0, cls_W, cls_b, logits_out, BB, 32, 2, 2, 0, 0);    // logits -> d_out
    logsoftmax2_ker<<<GRID(BB), TPB, 0, s>>>(logits_out, out);
    losses_ker<<<1, 32, 0, s>>>(ACC, out);
    #undef GRID
}